// NbModel_44624710205943
// MI455X (gfx1250) — compile-verified
//
#include <hip/hip_runtime.h>

typedef __attribute__((ext_vector_type(2))) float v2f;
typedef __attribute__((ext_vector_type(8))) float v8f;

#define GG    1024     // grid points (32*32)
#define KB    1024     // basis count
#define HD    256      // siren hidden dim
#define ROWS  3072     // B*3
#define NSTEP 100
#define MT    32       // rows of A per workgroup (two 16-row WMMA M-groups)

// ---------------------------------------------------------------------------
// Kernel 1: Siren basis evaluation (unchanged structure; WMMA f32 16x16x4).
// One block = 16 grid points, 256 threads = 8 waves.
// Writes Bas[k][p] (canonical, == basis_.T output) and BasT[p][k] (ws).
// ---------------------------------------------------------------------------
__global__ __launch_bounds__(256) void siren_kernel(
    const float* __restrict__ W0, const float* __restrict__ b0,
    const float* __restrict__ Wh, const float* __restrict__ bh,
    const float* __restrict__ Wl, const float* __restrict__ bl,
    float* __restrict__ bas_out,   // [K][P]
    float* __restrict__ basT)      // [P][K]
{
    __shared__ float hA[HD * 16];
    __shared__ float hB[HD * 16];

    const int tid  = threadIdx.x;
    const int lane = tid & 31;
    const int wave = tid >> 5;
    const int half = lane >> 4;
    const int ml   = lane & 15;
    const int p0   = blockIdx.x * 16;

    {   // layer 0: h = sin(30 * (plane @ W0 + b0)); thread t owns hidden k=t
        const int k    = tid;
        const float wx = W0[k];
        const float wy = W0[HD + k];
        const float bb = b0[k];
        #pragma unroll
        for (int m = 0; m < 16; ++m) {
            int p = p0 + m;
            float gi = -1.0f + 2.0f * (float)(p >> 5) * (1.0f / 31.0f);
            float gj = -1.0f + 2.0f * (float)(p & 31) * (1.0f / 31.0f);
            hA[k * 16 + m] = __sinf(30.0f * (gi * wx + gj * wy + bb));
        }
    }
    __syncthreads();

    float* hin  = hA;
    float* hout = hB;

    #pragma unroll 1
    for (int layer = 0; layer < 4; ++layer) {
        const float* W   = Wh + layer * HD * HD;
        const float* bhl = bh + layer * HD;
        const int n0 = wave * 32;
        v8f acc0 = {}, acc1 = {};
        #pragma unroll 4
        for (int k0 = 0; k0 < HD; k0 += 4) {
            int ka = k0 + 2 * half;
            v2f a   = { hin[ka * 16 + ml], hin[(ka + 1) * 16 + ml] };
            v2f bv0 = { W[ka * HD + n0 + ml],      W[(ka + 1) * HD + n0 + ml] };
            v2f bv1 = { W[ka * HD + n0 + 16 + ml], W[(ka + 1) * HD + n0 + 16 + ml] };
            acc0 = __builtin_amdgcn_wmma_f32_16x16x4_f32(false, a, false, bv0, (short)0, acc0, false, false);
            acc1 = __builtin_amdgcn_wmma_f32_16x16x4_f32(false, a, false, bv1, (short)0, acc1, false, false);
        }
        #pragma unroll
        for (int j = 0; j < 8; ++j) {
            int m = j + half * 8;
            hout[(n0 + ml) * 16 + m]      = __sinf(acc0[j] + bhl[n0 + ml]);
            hout[(n0 + 16 + ml) * 16 + m] = __sinf(acc1[j] + bhl[n0 + 16 + ml]);
        }
        __syncthreads();
        float* t = hin; hin = hout; hout = t;
    }

    #pragma unroll 1
    for (int t = 0; t < 4; ++t) {
        const int n0 = wave * 128 + t * 32;
        v8f acc0 = {}, acc1 = {};
        #pragma unroll 4
        for (int k0 = 0; k0 < HD; k0 += 4) {
            int ka = k0 + 2 * half;
            v2f a   = { hin[ka * 16 + ml], hin[(ka + 1) * 16 + ml] };
            v2f bv0 = { Wl[ka * KB + n0 + ml],      Wl[(ka + 1) * KB + n0 + ml] };
            v2f bv1 = { Wl[ka * KB + n0 + 16 + ml], Wl[(ka + 1) * KB + n0 + 16 + ml] };
            acc0 = __builtin_amdgcn_wmma_f32_16x16x4_f32(false, a, false, bv0, (short)0, acc0, false, false);
            acc1 = __builtin_amdgcn_wmma_f32_16x16x4_f32(false, a, false, bv1, (short)0, acc1, false, false);
        }
        #pragma unroll
        for (int j = 0; j < 8; ++j) {
            int m = j + half * 8;
            int p = p0 + m;
            int n = n0 + ml;
            float v0 = acc0[j] + bl[n];
            float v1 = acc1[j] + bl[n + 16];
            bas_out[n * GG + p]        = v0;
            bas_out[(n + 16) * GG + p] = v1;
            basT[p * KB + n]           = v0;
            basT[p * KB + n + 16]      = v1;
        }
    }
}

// ---------------------------------------------------------------------------
// Kernel 2: repack a [1024][1024] matrix into WMMA-B-fragment order:
//   dst_pair[q][c] = { src[2q][c], src[2q+1][c] }   (q = row pair, c = column)
// Flat dst index o: parity r = o&1, column c = (o>>1)&1023, pair q = o>>11.
// One global_load_b64 then fetches a lane's whole B fragment.
// ---------------------------------------------------------------------------
__global__ __launch_bounds__(256) void repack_kernel(
    const float* __restrict__ src, float* __restrict__ dst)
{
    for (int o = blockIdx.x * 2048 + threadIdx.x; o < (blockIdx.x + 1) * 2048; o += 256) {
        int r = o & 1;
        int c = (o >> 1) & (GG - 1);
        int q = o >> 11;
        dst[o] = src[(2 * q + r) * GG + c];
    }
}

// ---------------------------------------------------------------------------
// Kernel 3: 100 fused Adam steps per 32-row strip of A. 512 threads = 16 waves.
// Each wave owns a 64-column strip (4 N-tiles) and both 16-row M-groups.
// pass1: resid = A*Bas - x   (B fragments: b64 from BasP; resid -> LDS)
// pass2: g = (2/n)*resid*Bas^T (B fragments: b64 from BasTP); Adam update.
// Final: y = A*Bas -> d_out.
// ---------------------------------------------------------------------------
__global__ __launch_bounds__(512) void adam_kernel(
    const float* __restrict__ x,      // [ROWS][GG]
    const float* __restrict__ basP,   // pair-packed Bas   (pass-1 B operand)
    const float* __restrict__ basTP,  // pair-packed Bas^T (pass-2 B operand)
    float* __restrict__ m_ws,         // [ROWS][K]
    float* __restrict__ v_ws,         // [ROWS][K]
    float* __restrict__ y_out)        // [ROWS][GG]
{
    __shared__ float A_lds[KB * MT];  // [k][m], m = 0..31      (128 KB)
    __shared__ float R_lds[GG * MT];  // [p][m], m = 0..31      (128 KB)

    const int tid  = threadIdx.x;
    const int lane = tid & 31;
    const int wave = tid >> 5;
    const int half = lane >> 4;
    const int ml   = lane & 15;
    const int r0   = blockIdx.x * MT;
    const float two_over_n = 2.0f / 3145728.0f;
    const v2f* BP  = (const v2f*)basP;
    const v2f* BTP = (const v2f*)basTP;

    for (int i = tid; i < KB * MT; i += 512) A_lds[i] = 1.0f;
    for (int i = tid; i < MT * KB; i += 512) {
        int row = i >> 10, k = i & (KB - 1);
        m_ws[(r0 + row) * KB + k] = 0.0f;
        v_ws[(r0 + row) * KB + k] = 0.0f;
    }
    __syncthreads();

    float b1t = 1.0f, b2t = 1.0f;
    const int strip = wave * 64;      // this wave's 64-column strip

    #pragma unroll 1
    for (int step = 0; step < NSTEP; ++step) {
        b1t *= 0.9f; b2t *= 0.999f;
        const float bc1 = 1.0f / (1.0f - b1t);
        const float bc2 = 1.0f / (1.0f - b2t);

        // ---- pass 1: resid = A*Bas - x  (8 WMMA per k-step, shared B)
        {
            v8f acc[2][4];
            #pragma unroll
            for (int g = 0; g < 2; ++g)
                #pragma unroll
                for (int t = 0; t < 4; ++t) acc[g][t] = (v8f){};
            #pragma unroll 2
            for (int k0 = 0; k0 < KB; k0 += 4) {
                int ka = k0 + 2 * half;
                int kq = (k0 >> 1) + half;
                v2f a0 = { A_lds[ka * MT + ml],      A_lds[(ka + 1) * MT + ml] };
                v2f a1 = { A_lds[ka * MT + 16 + ml], A_lds[(ka + 1) * MT + 16 + ml] };
                #pragma unroll
                for (int t = 0; t < 4; ++t) {
                    v2f bv = BP[kq * GG + strip + t * 16 + ml];
                    acc[0][t] = __builtin_amdgcn_wmma_f32_16x16x4_f32(false, a0, false, bv, (short)0, acc[0][t], false, false);
                    acc[1][t] = __builtin_amdgcn_wmma_f32_16x16x4_f32(false, a1, false, bv, (short)0, acc[1][t], false, false);
                }
            }
            #pragma unroll
            for (int t = 0; t < 4; ++t)
                #pragma unroll
                for (int j = 0; j < 8; ++j) {
                    int m = j + half * 8;
                    int p = strip + t * 16 + ml;
                    R_lds[p * MT + m]      = acc[0][t][j] - x[(r0 + m) * GG + p];
                    R_lds[p * MT + 16 + m] = acc[1][t][j] - x[(r0 + 16 + m) * GG + p];
                }
        }
        __syncthreads();

        // ---- pass 2: g = (2/n)*resid*Bas^T ; Adam update of A/m/v
        {
            v8f acc[2][4];
            #pragma unroll
            for (int g = 0; g < 2; ++g)
                #pragma unroll
                for (int t = 0; t < 4; ++t) acc[g][t] = (v8f){};
            #pragma unroll 2
            for (int q0 = 0; q0 < GG; q0 += 4) {
                int qa = q0 + 2 * half;
                int qq = (q0 >> 1) + half;
                v2f a0 = { R_lds[qa * MT + ml],      R_lds[(qa + 1) * MT + ml] };
                v2f a1 = { R_lds[qa * MT + 16 + ml], R_lds[(qa + 1) * MT + 16 + ml] };
                #pragma unroll
                for (int t = 0; t < 4; ++t) {
                    v2f bv = BTP[qq * KB + strip + t * 16 + ml];
                    acc[0][t] = __builtin_amdgcn_wmma_f32_16x16x4_f32(false, a0, false, bv, (short)0, acc[0][t], false, false);
                    acc[1][t] = __builtin_amdgcn_wmma_f32_16x16x4_f32(false, a1, false, bv, (short)0, acc[1][t], false, false);
                }
            }
            #pragma unroll
            for (int t = 0; t < 4; ++t)
                #pragma unroll
                for (int j = 0; j < 8; ++j)
                    #pragma unroll
                    for (int mg = 0; mg < 2; ++mg) {
                        int m = j + half * 8 + mg * 16;
                        int k = strip + t * 16 + ml;
                        float g  = two_over_n * acc[mg][t][j];
                        int idx  = (r0 + m) * KB + k;
                        float mm = 0.9f   * m_ws[idx] + 0.1f   * g;
                        float vv = 0.999f * v_ws[idx] + 0.001f * g * g;
                        m_ws[idx] = mm;
                        v_ws[idx] = vv;
                        A_lds[k * MT + m] -= 0.1f * (mm * bc1)
                                           / (__builtin_sqrtf(vv * bc2) + 1e-8f);
                    }
        }
        __syncthreads();
    }

    // ---- final y = A * Bas
    {
        v8f acc[2][4];
        #pragma unroll
        for (int g = 0; g < 2; ++g)
            #pragma unroll
            for (int t = 0; t < 4; ++t) acc[g][t] = (v8f){};
        #pragma unroll 2
        for (int k0 = 0; k0 < KB; k0 += 4) {
            int ka = k0 + 2 * half;
            int kq = (k0 >> 1) + half;
            v2f a0 = { A_lds[ka * MT + ml],      A_lds[(ka + 1) * MT + ml] };
            v2f a1 = { A_lds[ka * MT + 16 + ml], A_lds[(ka + 1) * MT + 16 + ml] };
            #pragma unroll
            for (int t = 0; t < 4; ++t) {
                v2f bv = BP[kq * GG + strip + t * 16 + ml];
                acc[0][t] = __builtin_amdgcn_wmma_f32_16x16x4_f32(false, a0, false, bv, (short)0, acc[0][t], false, false);
                acc[1][t] = __builtin_amdgcn_wmma_f32_16x16x4_f32(false, a1, false, bv, (short)0, acc[1][t], false, false);
            }
        }
        #pragma unroll
        for (int t = 0; t < 4; ++t)
            #pragma unroll
            for (int j = 0; j < 8; ++j) {
                int m = j + half * 8;
                int p = strip + t * 16 + ml;
                y_out[(r0 + m) * GG + p]      = acc[0][t][j];
                y_out[(r0 + 16 + m) * GG + p] = acc[1][t][j];
            }
    }
}

// ---------------------------------------------------------------------------
extern "C" void kernel_launch(void* const* d_in, const int* in_sizes, int n_in,
                              void* d_out, int out_size, void* d_ws, size_t ws_size,
                              hipStream_t stream) {
    const float* x  = (const float*)d_in[0];
    const float* W0 = (const float*)d_in[1];
    const float* b0 = (const float*)d_in[2];
    const float* Wh = (const float*)d_in[3];
    const float* bh = (const float*)d_in[4];
    const float* Wl = (const float*)d_in[5];
    const float* bl = (const float*)d_in[6];

    float* y_out   = (float*)d_out;                 // [3072][1024]
    float* bas_out = y_out + (size_t)ROWS * GG;     // basis_.T, [1024][1024]

    float* basT  = (float*)d_ws;                    // canonical Bas^T (4 MB)
    float* basP  = basT  + (size_t)GG * KB;         // pair-packed Bas (4 MB)
    float* basTP = basP  + (size_t)GG * KB;         // pair-packed Bas^T (4 MB)
    float* m_ws  = basTP + (size_t)GG * KB;         // 12 MB
    float* v_ws  = m_ws  + (size_t)ROWS * KB;       // 12 MB

    siren_kernel<<<GG / 16, 256, 0, stream>>>(W0, b0, Wh, bh, Wl, bl, bas_out, basT);
    repack_kernel<<<1024, 256, 0, stream>>>(bas_out, basP);
    repack_kernel<<<1024, 256, 0, stream>>>(basT, basTP);
    adam_kernel<<<ROWS / MT, 512, 0, stream>>>(x, basP, basTP, m_ws, v_ws, y_out);
}